// ImageEncoder_41944650613092
// MI455X (gfx1250) — compile-verified
//
#include <hip/hip_runtime.h>
#include <hip/hip_bf16.h>

namespace {
constexpr int kSLEN = 208;
constexpr int kPT   = 8;
constexpr int kSTEP = 2;
constexpr int kNT   = (kSLEN - kPT) / kSTEP + 1;   // 101
constexpr int kP    = kNT * kNT;                   // 10201 tiles
constexpr int kB    = 16;
constexpr int kS    = 100;
constexpr int kNF   = 5;
constexpr int kBP   = kB * kP;                     // 163216
constexpr int kThreads = 256;
}

// gfx1250 async global->LDS staging (ASYNCcnt path). Guarded so the file
// compiles on either toolchain; fallback is a plain LDS copy.
#if defined(__HIP_DEVICE_COMPILE__) && defined(__gfx1250__) && \
    __has_builtin(__builtin_amdgcn_global_load_async_to_lds_b32)
#define USE_ASYNC_LDS 1
#else
#define USE_ASYNC_LDS 0
#endif

#if USE_ASYNC_LDS
typedef __attribute__((address_space(1))) int* gbl_i32_ptr;
typedef __attribute__((address_space(3))) int* lds_i32_ptr;
#endif

__global__ __launch_bounds__(kThreads) void tile_encoder_kernel(
    const float* __restrict__ locs,    // (B,S,2)
    const float* __restrict__ fluxes,  // (B,S,5)
    float* __restrict__ out)           // [BP | 4*BP | 10*BP | 2*BP] floats
{
    __shared__ float sloc[kS * 2];     // 200 floats
    __shared__ float sflx[kS * kNF];   // 500 floats

    const int tid = threadIdx.x;
    const int b   = blockIdx.y;
    const int p   = blockIdx.x * kThreads + tid;

    const float* __restrict__ gl = locs   + (size_t)b * (kS * 2);
    const float* __restrict__ gf = fluxes + (size_t)b * (kS * kNF);

#if USE_ASYNC_LDS
    for (int i = tid; i < kS * 2; i += kThreads) {
        __builtin_amdgcn_global_load_async_to_lds_b32(
            (gbl_i32_ptr)(gl + i), (lds_i32_ptr)(sloc + i), 0, 0);
    }
    for (int i = tid; i < kS * kNF; i += kThreads) {
        __builtin_amdgcn_global_load_async_to_lds_b32(
            (gbl_i32_ptr)(gf + i), (lds_i32_ptr)(sflx + i), 0, 0);
    }
#if __has_builtin(__builtin_amdgcn_s_wait_asynccnt)
    __builtin_amdgcn_s_wait_asynccnt(0);
#else
    asm volatile("s_wait_asynccnt 0" ::: "memory");
#endif
    __syncthreads();
#else
    for (int i = tid; i < kS * 2; i += kThreads)   sloc[i] = gl[i];
    for (int i = tid; i < kS * kNF; i += kThreads) sflx[i] = gf[i];
    __syncthreads();
#endif

    if (p >= kP) return;

    // tile coords: dim0 = (p / 101)*2, dim1 = (p % 101)*2
    const int ty = p / kNT;
    const int tx = p - ty * kNT;
    const float l0 = (float)(ty * kSTEP) + 2.5f;   // left  = tc + EDGE - 0.5
    const float l1 = (float)(tx * kSTEP) + 2.5f;
    const float r0 = l0 + 2.0f;                    // right = tc + 4.5 (exact)
    const float r1 = l1 + 2.0f;

    int cnt = 0, s0 = -1, s1 = -1;
    for (int s = 0; s < kS; ++s) {
        const float lp0 = sloc[2 * s]     * 207.0f;
        const float lp1 = sloc[2 * s + 1] * 207.0f;
        const bool on = (lp0 > l0) & (lp0 < r0) & (lp0 != 0.0f) &
                        (lp1 > l1) & (lp1 < r1) & (lp1 != 0.0f);
        if (on) {
            if (cnt == 0)      s0 = s;
            else if (cnt == 1) s1 = s;
            ++cnt;
        }
    }

    const int q = b * kP + p;

    // output 0: n_sources (clamped), stored as float
    out[q] = (float)(cnt < 2 ? cnt : 2);

    // slots: on-hits get (lp-left)*0.5 (positive => relu no-op), misses are 0
    float4 lo = make_float4(0.f, 0.f, 0.f, 0.f);
    float  fx[10];
#pragma unroll
    for (int i = 0; i < 10; ++i) fx[i] = 0.f;
    float io0 = 0.f, io1 = 0.f;

    if (s0 >= 0) {
        const float a = sloc[2 * s0] * 207.0f;
        const float c = sloc[2 * s0 + 1] * 207.0f;
        lo.x = (a - l0) * 0.5f;
        lo.y = (c - l1) * 0.5f;
#pragma unroll
        for (int f = 0; f < kNF; ++f) fx[f] = sflx[s0 * kNF + f];
        io0 = 1.f;
    }
    if (s1 >= 0) {
        const float a = sloc[2 * s1] * 207.0f;
        const float c = sloc[2 * s1 + 1] * 207.0f;
        lo.z = (a - l0) * 0.5f;
        lo.w = (c - l1) * 0.5f;
#pragma unroll
        for (int f = 0; f < kNF; ++f) fx[kNF + f] = sflx[s1 * kNF + f];
        io1 = 1.f;
    }

    // output 1: tile_locs (BP,2,2) — 16B aligned (base BP floats, q*16B)
    *(float4*)(out + kBP + (size_t)q * 4) = lo;

    // output 2: tile_fluxes (BP,2,5) — 8B aligned (base 5*BP floats = 20BP bytes, q*40B)
    float2* f2 = (float2*)(out + (size_t)5 * kBP + (size_t)q * 10);
#pragma unroll
    for (int k = 0; k < 5; ++k) f2[k] = make_float2(fx[2 * k], fx[2 * k + 1]);

    // output 3: tile_is_on (BP,2) — 8B aligned (base 15*BP floats, q*8B)
    *(float2*)(out + (size_t)15 * kBP + (size_t)q * 2) = make_float2(io0, io1);
}

extern "C" void kernel_launch(void* const* d_in, const int* in_sizes, int n_in,
                              void* d_out, int out_size, void* d_ws, size_t ws_size,
                              hipStream_t stream) {
    const float* locs   = (const float*)d_in[0];   // 16*100*2
    const float* fluxes = (const float*)d_in[1];   // 16*100*5
    float* out = (float*)d_out;                    // 17 * 163216 floats

    dim3 grid((kP + kThreads - 1) / kThreads, kB); // (40, 16)
    tile_encoder_kernel<<<grid, kThreads, 0, stream>>>(locs, fluxes, out);
}